// DoublyStochasticButterfly_16466904613530
// MI455X (gfx1250) — compile-verified
//
#include <hip/hip_runtime.h>

// Doubly stochastic butterfly, algebraically collapsed:
//   s = x0+x1 invariant under each doubly-stochastic 2x2 mix;
//   d = x0-x1 scales by (1-2w) each depth step
//   => out0 = (s + d*P)/2, out1 = (s - d*P)/2 with P[j] = prod_i (1-2*params[j,i])
// Pure HBM stream: 256 MB traffic -> ~11.5us floor at 23.3 TB/s; ~0.27 GFLOP.
// WMMA inapplicable (per-column 2x2 diagonal mixes, not a shared matrix).
// NT temporal hints on the single-use X/out streams keep L2 clean.

#define BWIDTH 1024
#define BHALF  512
#define BDEPTH 32
#define BBATCH 32768
#define ROWS_PER_BLOCK 8

typedef float v4f __attribute__((ext_vector_type(4)));

// ---- Kernel 1: reduce params (512x32) -> P (512) --------------------------
__global__ void __launch_bounds__(256)
butterfly_prod_kernel(const float* __restrict__ params, float* __restrict__ P) {
    int j = blockIdx.x * blockDim.x + threadIdx.x;
    if (j < BHALF) {
        const v4f* pv = reinterpret_cast<const v4f*>(params + (size_t)j * BDEPTH);
        float prod = 1.0f;
#pragma unroll
        for (int k = 0; k < BDEPTH / 4; ++k) {
            v4f v = pv[k];
            prod *= (1.0f - 2.0f * v.x);
            prod *= (1.0f - 2.0f * v.y);
            prod *= (1.0f - 2.0f * v.z);
            prod *= (1.0f - 2.0f * v.w);
        }
        P[j] = prod;
    }
}

// ---- Kernel 2: streaming apply -------------------------------------------
__global__ void __launch_bounds__(256)
butterfly_apply_kernel(const float* __restrict__ X,
                       const float* __restrict__ P,
                       float* __restrict__ out) {
    __shared__ __align__(16) float Pl[BHALF];

    const int tid = threadIdx.x;

    // Stage the 2KB P table into LDS via CDNA5 async copy (ASYNCcnt path).
    // 128 lanes x B128 = 512 floats.
    if (tid < BHALF / 4) {
        unsigned lds_off = (unsigned)(size_t)(&Pl[tid * 4]);   // low 32 bits of flat LDS
                                                               // addr == LDS byte offset
        unsigned long long gaddr = (unsigned long long)(P + tid * 4);
        asm volatile("global_load_async_to_lds_b128 %0, %1, off"
                     :
                     : "v"(lds_off), "v"(gaddr)
                     : "memory");
    }
    asm volatile("s_wait_asynccnt 0" ::: "memory");
    __syncthreads();

    const int cg = tid & (BHALF / 4 - 1);  // column group: 0..127 (4 pair-columns each)
    const int rl = tid >> 7;               // row lane within block iteration: 0..1

    const v4f p4 = *reinterpret_cast<const v4f*>(&Pl[cg * 4]);  // ds_load_b128, reused

    size_t rowbase = ((size_t)blockIdx.x * ROWS_PER_BLOCK + (size_t)rl) * BWIDTH
                     + (size_t)cg * 4;

#pragma unroll
    for (int r = 0; r < ROWS_PER_BLOCK / 2; ++r) {
        size_t idx = rowbase + (size_t)(2 * r) * BWIDTH;

        if (r + 1 < ROWS_PER_BLOCK / 2) {
            // CDNA5 global_prefetch_b8 for the next row-pair this thread touches
            __builtin_prefetch(X + idx + 2 * BWIDTH, 0, 0);
            __builtin_prefetch(X + idx + 2 * BWIDTH + BHALF, 0, 0);
        }

        // Single-use stream: non-temporal loads (th:TH_LOAD_NT)
        v4f x0 = __builtin_nontemporal_load(reinterpret_cast<const v4f*>(X + idx));
        v4f x1 = __builtin_nontemporal_load(reinterpret_cast<const v4f*>(X + idx + BHALF));

        v4f o0, o1;
        {
            float s, d;
            s = 0.5f * (x0.x + x1.x); d = 0.5f * (x0.x - x1.x) * p4.x; o0.x = s + d; o1.x = s - d;
            s = 0.5f * (x0.y + x1.y); d = 0.5f * (x0.y - x1.y) * p4.y; o0.y = s + d; o1.y = s - d;
            s = 0.5f * (x0.z + x1.z); d = 0.5f * (x0.z - x1.z) * p4.z; o0.z = s + d; o1.z = s - d;
            s = 0.5f * (x0.w + x1.w); d = 0.5f * (x0.w - x1.w) * p4.w; o0.w = s + d; o1.w = s - d;
        }

        // Single-use stream: non-temporal stores (th:TH_STORE_NT)
        __builtin_nontemporal_store(o0, reinterpret_cast<v4f*>(out + idx));
        __builtin_nontemporal_store(o1, reinterpret_cast<v4f*>(out + idx + BHALF));
    }
}

extern "C" void kernel_launch(void* const* d_in, const int* in_sizes, int n_in,
                              void* d_out, int out_size, void* d_ws, size_t ws_size,
                              hipStream_t stream) {
    const float* X      = (const float*)d_in[0];   // (32768, 1024) f32
    const float* params = (const float*)d_in[1];   // (512, 32)    f32
    float*       out    = (float*)d_out;           // (32768, 1024) f32
    float*       P      = (float*)d_ws;            // 512 f32 scratch

    butterfly_prod_kernel<<<(BHALF + 255) / 256, 256, 0, stream>>>(params, P);
    butterfly_apply_kernel<<<BBATCH / ROWS_PER_BLOCK, 256, 0, stream>>>(X, P, out);
}